// LambdaLayer_31421980738301
// MI455X (gfx1250) — compile-verified
//
#include <hip/hip_runtime.h>
#include <hip/hip_bf16.h>
#include <math.h>

typedef __attribute__((ext_vector_type(2))) float v2f;
typedef __attribute__((ext_vector_type(8))) float v8f;

#define NH    4
#define DK    16
#define DV    64
#define BATCH 32
#define CIN   256
#define MM    1024
#define EPSV  1e-5f

__device__ __forceinline__ v8f wmma_f32_k4(v2f a, v2f b, v8f c) {
  // D(16x16,f32) = A(16x4,f32) x B(4x16,f32) + C
  return __builtin_amdgcn_wmma_f32_16x16x4_f32(false, a, false, b, (short)0, c,
                                               false, false);
}

// ---------------------------------------------------------------------------
// Kernel A: per-batch GEMM qkv = W[144,256] @ x[256,1024], BN fused.
// 9 waves/block (one per 16-channel o-tile), grid = (64 m-tiles, 32 batches).
// The shared x[256x16] tile is staged into LDS ONCE per block via gfx1250
// async global->LDS copies (ASYNCcnt), then all 9 waves feed their WMMAs
// from LDS. o-tiles 0..3 (q) and 5..8 (v) compute the transposed product so
// qT/vT (m-major) stores are coalesced; o-tile 4 (k) computes the normal
// orientation so ksm (channel-major) stores are coalesced in m.
// ---------------------------------------------------------------------------
__global__ __launch_bounds__(288) void qkv_gemm_bn_kernel(
    const float* __restrict__ x, const float* __restrict__ Wq,
    const float* __restrict__ qg, const float* __restrict__ qb,
    const float* __restrict__ qm, const float* __restrict__ qv,
    const float* __restrict__ vg, const float* __restrict__ vb,
    const float* __restrict__ vm, const float* __restrict__ vv,
    float* __restrict__ qT, float* __restrict__ vT, float* __restrict__ ksm)
{
  __shared__ float xs[CIN * 16];         // 16 KB: xs[c][m_local]
  const int mtile = blockIdx.x;          // 0..63
  const int b     = blockIdx.y;          // 0..31
  const int tid   = threadIdx.x;
  const int otile = tid >> 5;            // 0..8 (wave id)
  const int lane  = tid & 31;
  const int col   = lane & 15;
  const int hi    = lane >> 4;
  const int kb    = hi << 1;
  const int m0    = mtile << 4;
  const float* xb = x + (size_t)b * CIN * MM;

  // ---- async-stage x tile: 1024 x b128 lane-transfers (16 B each) ----
  for (int i = tid; i < CIN * 4; i += 288) {
    const int c = i >> 2;
    const int g = (i & 3) << 2;                     // 4-float group within row
    const float* gp = xb + (size_t)c * MM + m0 + g; // 16B-aligned
    unsigned lds = (unsigned)(uintptr_t)(&xs[c * 16 + g]);
    asm volatile("global_load_async_to_lds_b128 %0, %1, off"
                 :: "v"(lds), "v"(gp) : "memory");
  }
  asm volatile("s_wait_asynccnt 0x0" ::: "memory");
  __syncthreads();

  v8f acc = {0.f, 0.f, 0.f, 0.f, 0.f, 0.f, 0.f, 0.f};

  if (otile == 4) {
    // k tile: D rows = channels 64..79, D cols = m
    const int orow = 64 + col;
    const int mcol = m0 + col;
    const float* wrow = Wq + orow * CIN;
    for (int c0 = 0; c0 < CIN; c0 += 4) {
      v2f a = *(const v2f*)(wrow + c0 + kb);         // A[orow][c0+kb .. +1]
      v2f bv;
      bv.x = xs[(c0 + kb)     * 16 + col];           // B[c][m] from LDS
      bv.y = xs[(c0 + kb + 1) * 16 + col];
      acc = wmma_f32_k4(a, bv, acc);
    }
    float* kout = ksm + (size_t)b * DK * MM;
#pragma unroll
    for (int r = 0; r < 8; ++r) {
      int ch = r + (hi << 3);                        // 0..15
      kout[ch * MM + mcol] = acc[r];                 // coalesced in m
    }
  } else {
    // q / v tiles: transposed product, D rows = m, D cols = channel
    const int ocol = (otile << 4) + col;
    const float* wcol = Wq + ocol * CIN;
    for (int c0 = 0; c0 < CIN; c0 += 4) {
      v2f a;
      a.x = xs[(c0 + kb)     * 16 + col];            // A[m][c] from LDS
      a.y = xs[(c0 + kb + 1) * 16 + col];
      v2f bv = *(const v2f*)(wcol + c0 + kb);        // B[c][ocol] = W[ocol][c]
      acc = wmma_f32_k4(a, bv, acc);
    }
    float scl, sft;
    float* dst;
    int dcol;
    if (otile < 4) {                                 // q channels 0..63
      scl = qg[ocol] * rsqrtf(qv[ocol] + EPSV);
      sft = qb[ocol] - qm[ocol] * scl;
      dst = qT; dcol = ocol;
    } else {                                         // v channels 80..143
      int vi = ocol - 80;
      scl = vg[vi] * rsqrtf(vv[vi] + EPSV);
      sft = vb[vi] - vm[vi] * scl;
      dst = vT; dcol = vi;
    }
    dst += (size_t)b * MM * DV;
#pragma unroll
    for (int r = 0; r < 8; ++r) {
      int m = m0 + r + (hi << 3);
      dst[(size_t)m * DV + dcol] = acc[r] * scl + sft;  // coalesced in channel
    }
  }
}

// ---------------------------------------------------------------------------
// Kernel B: softmax over M=1024 for each of the 32*16 k rows (in place).
// ---------------------------------------------------------------------------
__global__ __launch_bounds__(256) void k_softmax_kernel(float* __restrict__ ksm)
{
  const int row = blockIdx.x;                        // b*16 + j
  float* p = ksm + (size_t)row * MM;
  __shared__ float red[256];
  const int t = threadIdx.x;
  float v0 = p[t], v1 = p[t + 256], v2 = p[t + 512], v3 = p[t + 768];
  float mx = fmaxf(fmaxf(v0, v1), fmaxf(v2, v3));
  red[t] = mx; __syncthreads();
  for (int s = 128; s > 0; s >>= 1) {
    if (t < s) red[t] = fmaxf(red[t], red[t + s]);
    __syncthreads();
  }
  mx = red[0]; __syncthreads();
  v0 = expf(v0 - mx); v1 = expf(v1 - mx); v2 = expf(v2 - mx); v3 = expf(v3 - mx);
  red[t] = v0 + v1 + v2 + v3; __syncthreads();
  for (int s = 128; s > 0; s >>= 1) {
    if (t < s) red[t] += red[t + s];
    __syncthreads();
  }
  float inv = 1.0f / red[0];
  p[t] = v0 * inv; p[t + 256] = v1 * inv; p[t + 512] = v2 * inv; p[t + 768] = v3 * inv;
}

// ---------------------------------------------------------------------------
// Kernel C: content_lam[b][16][64] = ksm[b][16][1024] @ vT[b][1024][64].
// One block per batch, 4 waves (one per 16-col v tile), 256 WMMA K-steps.
// ---------------------------------------------------------------------------
__global__ __launch_bounds__(128) void content_lam_kernel(
    const float* __restrict__ ksm, const float* __restrict__ vT,
    float* __restrict__ clam)
{
  const int b    = blockIdx.x;
  const int nt   = threadIdx.x >> 5;                 // v tile 0..3
  const int lane = threadIdx.x & 31;
  const int col  = lane & 15;
  const int hi   = lane >> 4;
  const int kb   = hi << 1;
  const float* K = ksm + (size_t)b * DK * MM;
  const float* V = vT  + (size_t)b * MM * DV;
  const float* krow = K + col * MM;
  v8f acc = {0.f, 0.f, 0.f, 0.f, 0.f, 0.f, 0.f, 0.f};
  for (int m0 = 0; m0 < MM; m0 += 4) {
    v2f a = *(const v2f*)(krow + m0 + kb);           // A[j][m], j = col
    v2f bv;
    bv.x = V[(m0 + kb)     * DV + (nt << 4) + col];  // B[m][v]
    bv.y = V[(m0 + kb + 1) * DV + (nt << 4) + col];
    acc = wmma_f32_k4(a, bv, acc);
  }
  float* C = clam + (size_t)b * DK * DV;
#pragma unroll
  for (int r = 0; r < 8; ++r)
    C[(r + (hi << 3)) * DV + (nt << 4) + col] = acc[r];
}

// ---------------------------------------------------------------------------
// Kernel D: per position m, build L = conv7x7(v) (+lam_b +content_lam) as an
// implicit GEMM with WMMA (A = 16x52 zero-padded weights, B = 52x64 gathered
// neighbors), then out[n][v] = q[n][:] . L[:][v] via FMAs + ds_swizzle SWAPX16
// half-wave reduction, stored in the reference's permuted output layout.
// 4 waves/block, one position per wave.
// ---------------------------------------------------------------------------
__global__ __launch_bounds__(128) void lambda_out_kernel(
    const float* __restrict__ qT, const float* __restrict__ vT,
    const float* __restrict__ clam, const float* __restrict__ lw,
    const float* __restrict__ lb, float* __restrict__ out)
{
  const int wave = threadIdx.x >> 5;
  const int lane = threadIdx.x & 31;
  const int col  = lane & 15;
  const int hi   = lane >> 4;
  const int gm   = (blockIdx.x << 2) + wave;         // 0..32767
  const int b    = gm >> 10;
  const int m    = gm & 1023;
  const int mh   = m >> 5, mw = m & 31;
  const float* V = vT + (size_t)b * MM * DV;

  v8f acc[4];
  v8f z = {0.f, 0.f, 0.f, 0.f, 0.f, 0.f, 0.f, 0.f};
#pragma unroll
  for (int t = 0; t < 4; ++t) acc[t] = z;

  for (int s = 0; s < 13; ++s) {                     // K = 52 (49 real taps)
    const int k0 = (s << 2) + (hi << 1);
    v2f a;
    a.x = (k0     < 49) ? lw[col * 49 + k0]     : 0.f;
    a.y = (k0 + 1 < 49) ? lw[col * 49 + k0 + 1] : 0.f;

    int nb0 = 0, nb1 = 0;
    bool ok0 = false, ok1 = false;
    {
      int di = k0 / 7 - 3, dj = k0 % 7 - 3;
      int h = mh + di, w = mw + dj;
      ok0 = (k0 < 49) && (h >= 0) && (h < 32) && (w >= 0) && (w < 32);
      nb0 = h * 32 + w;
      int k1 = k0 + 1;
      int di1 = k1 / 7 - 3, dj1 = k1 % 7 - 3;
      int h1 = mh + di1, w1 = mw + dj1;
      ok1 = (k1 < 49) && (h1 >= 0) && (h1 < 32) && (w1 >= 0) && (w1 < 32);
      nb1 = h1 * 32 + w1;
    }
#pragma unroll
    for (int t = 0; t < 4; ++t) {
      v2f bv;
      bv.x = ok0 ? V[nb0 * DV + (t << 4) + col] : 0.f;
      bv.y = ok1 ? V[nb1 * DV + (t << 4) + col] : 0.f;
      acc[t] = wmma_f32_k4(a, bv, acc[t]);
    }
  }

  // L = pos_conv + lam_b + content_lam  (lane holds rows r+8*hi, col t*16+col)
  const float* CL = clam + (size_t)b * DK * DV;
  float L[4][8];
#pragma unroll
  for (int t = 0; t < 4; ++t)
#pragma unroll
    for (int r = 0; r < 8; ++r) {
      int row = r + (hi << 3);
      L[t][r] = acc[t][r] + CL[row * DV + (t << 4) + col] + lb[row];
    }

  __shared__ float qs[4][64];
  const float* Q = qT + ((size_t)b * MM + m) * DV;
  qs[wave][2 * lane]     = Q[2 * lane];
  qs[wave][2 * lane + 1] = Q[2 * lane + 1];
  __syncthreads();

  const int mhi2 = m >> 8;       // top 2 bits of m -> channel offset
  const int mlo  = m & 255;
  float* ob = out + (size_t)b * CIN * MM;
#pragma unroll
  for (int t = 0; t < 4; ++t) {
    const int c = (t << 4) + col;                    // v index 0..63
#pragma unroll
    for (int n = 0; n < 4; ++n) {
      float p = 0.f;
#pragma unroll
      for (int r = 0; r < 8; ++r)
        p += qs[wave][(n << 4) + r + (hi << 3)] * L[t][r];
      // SWAPX16: lane l <-> l^16 combine (rows 0-7 half + rows 8-15 half)
      int other = __builtin_amdgcn_ds_swizzle(__float_as_int(p), 0x401F);
      float tot = p + __int_as_float(other);
      if ((n >> 1) == hi)   // hi=0 writes n=0,1 ; hi=1 writes n=2,3
        ob[(size_t)(4 * c + mhi2) * MM + 4 * mlo + n] = tot;
    }
  }
}

// ---------------------------------------------------------------------------
extern "C" void kernel_launch(void* const* d_in, const int* in_sizes, int n_in,
                              void* d_out, int out_size, void* d_ws, size_t ws_size,
                              hipStream_t stream)
{
  const float* x  = (const float*)d_in[0];
  const float* Wq = (const float*)d_in[1];
  const float* qg = (const float*)d_in[2];
  const float* qb = (const float*)d_in[3];
  const float* qm = (const float*)d_in[4];
  const float* qv = (const float*)d_in[5];
  const float* vg = (const float*)d_in[6];
  const float* vb = (const float*)d_in[7];
  const float* vm = (const float*)d_in[8];
  const float* vv = (const float*)d_in[9];
  const float* lw = (const float*)d_in[10];
  const float* lb = (const float*)d_in[11];
  float* out = (float*)d_out;

  // workspace layout (floats): qT[32][1024][64], vT[32][1024][64],
  // ksm[32][16][1024], clam[32][16][64]  -> ~18.2 MB total
  float* ws   = (float*)d_ws;
  float* qT   = ws;
  float* vT   = qT  + (size_t)BATCH * MM * DV;
  float* ksm  = vT  + (size_t)BATCH * MM * DV;
  float* clam = ksm + (size_t)BATCH * DK * MM;

  qkv_gemm_bn_kernel<<<dim3(64, BATCH), 288, 0, stream>>>(
      x, Wq, qg, qb, qm, qv, vg, vb, vm, vv, qT, vT, ksm);
  k_softmax_kernel<<<BATCH * DK, 256, 0, stream>>>(ksm);
  content_lam_kernel<<<BATCH, 128, 0, stream>>>(ksm, vT, clam);
  lambda_out_kernel<<<(BATCH * MM) / 4, 128, 0, stream>>>(qT, vT, clam, lw, lb, out);
}